// ConvCaps_29970281791751
// MI455X (gfx1250) — compile-verified
//
#include <hip/hip_runtime.h>
#include <math.h>

// ---- problem constants (from reference) ----
#define BDIM   32      // input capsule types
#define CDIM   32      // output capsule types
#define PSIZE  16      // 4x4 pose
#define NIN    288     // K*K*B = 9*32
#define NPOS   128     // 2 * 8 * 8 spatial positions
#define VCOLS  512     // C * PSIZE
#define EMITERS 3
#define EPS    1e-6f
#define LAMBDA 1e-3f
#define LN_2PI 1.8378770664093453f

typedef __attribute__((ext_vector_type(2))) float v2f;
typedef __attribute__((ext_vector_type(8))) float v8f;

// -------------------------------------------------------------------------
// Kernel 1: patch extraction with the reference's reshape scramble.
// patches[b,y,x,k2,ch2] = xp[b, 2y+kh, 2x+kw, ch] where
//   f = k2*544 + ch2 ; ch = f/9 ; k = f%9 ; kh=k/3 ; kw=k%3   (pad=1)
// p_in[n,i,p]  <- k2 = (i*16+p)/512, ch2 = (i*16+p)%512
// a_in[n,i]    <- k2 = i/32,         ch2 = 512 + i%32
// -------------------------------------------------------------------------
__global__ void capsule_extract_kernel(const float* __restrict__ x,
                                       float* __restrict__ p_in,
                                       float* __restrict__ a_in) {
  const int NP = NPOS * NIN * PSIZE;   // 589824
  const int NA = NPOS * NIN;           // 36864
  int tid = blockIdx.x * blockDim.x + threadIdx.x;
  if (tid >= NP + NA) return;

  int n, k2, ch2, outIdx;
  float* dst;
  if (tid < NP) {
    int p    = tid & 15;
    int rest = tid >> 4;          // n*NIN + i
    int i    = rest % NIN;
    n        = rest / NIN;
    int g    = i * 16 + p;
    k2  = g >> 9;                 // g / 512
    ch2 = g & 511;
    dst = p_in; outIdx = tid;
  } else {
    int t2 = tid - NP;            // n*NIN + i
    int i  = t2 % NIN;
    n      = t2 / NIN;
    k2  = i >> 5;
    ch2 = 512 + (i & 31);
    dst = a_in; outIdx = t2;
  }
  int f  = k2 * 544 + ch2;
  int ch = f / 9;
  int k  = f - ch * 9;
  int kh = k / 3, kw = k - kh * 3;
  int b  = n >> 6;
  int oy = (n >> 3) & 7;
  int ox = n & 7;
  int iy = oy * 2 - 1 + kh;
  int ix = ox * 2 - 1 + kw;
  float val = 0.0f;
  if (iy >= 0 && iy < 16 && ix >= 0 && ix < 16)
    val = x[((b * 16 + iy) * 16 + ix) * 544 + ch];
  dst[outIdx] = val;
}

// -------------------------------------------------------------------------
// Kernel 2: vote GEMM with V_WMMA_F32_16X16X4_F32.
// For each input capsule i: V_i (128x512) = P_i (128x16) @ W_i (16x512),
// W_i[p][c*16+q] = weights[((i*32+c)*16+p)*16+q].
// One wave per 16x16 tile; K=16 -> 4 chained WMMAs.
// Tile nt maps exactly to output capsule c (N tile width == q stride == 16).
// -------------------------------------------------------------------------
__global__ __launch_bounds__(256)
void capsule_vote_wmma_kernel(const float* __restrict__ w,
                              const float* __restrict__ p_in,
                              float* __restrict__ v) {
  int lane = threadIdx.x & 31;
  int wave = blockIdx.x * 8 + (threadIdx.x >> 5);
  int nt = wave & 31;          // N tile == output capsule c
  int mt = (wave >> 5) & 7;    // M tile (spatial positions / 16)
  int i  = wave >> 8;          // input capsule (grid sized exactly: 288*256 waves)
  if (i >= NIN) return;        // wave-uniform; keeps EXEC all-1s for WMMA

  int hl  = lane >> 4;         // half-wave select
  int l15 = lane & 15;

  // A row for this lane (same for both halves), B column q = l15.
  const float* Ap = p_in + ((mt * 16 + l15) * NIN + i) * PSIZE;
  const float* Wp = w + (i * CDIM + nt) * (PSIZE * PSIZE) + l15;

  v8f acc = {0.f, 0.f, 0.f, 0.f, 0.f, 0.f, 0.f, 0.f};
#pragma unroll
  for (int kc = 0; kc < 4; ++kc) {
    int kb = kc * 4 + hl * 2;       // ISA 16x4 f32 layout: lanes<16 K={k0,k0+1}, lanes>=16 K={k0+2,k0+3}
    v2f af, bf;
    af.x = Ap[kb];
    af.y = Ap[kb + 1];
    bf.x = Wp[kb * 16];             // B 4x16: rows striped across lanes, symmetric half-split
    bf.y = Wp[(kb + 1) * 16];
    acc = __builtin_amdgcn_wmma_f32_16x16x4_f32(
        /*neg_a=*/false, af, /*neg_b=*/false, bf,
        /*c_mod=*/(short)0, acc, /*reuse_a=*/false, /*reuse_b=*/false);
  }

  // D layout: VGPR r -> rows M=r (lanes 0-15) and M=r+8 (lanes 16-31), N=l15.
  int col = nt * 16 + l15;
#pragma unroll
  for (int r = 0; r < 8; ++r) {
    int row = mt * 16 + r + hl * 8;
    v[(row * NIN + i) * VCOLS + col] = acc[r];
  }
}

// -------------------------------------------------------------------------
// Kernel 3: EM routing. One block (256 threads) per spatial position.
// All routing state in LDS; v[n] (288x512, 576 KB) streamed from L2.
// -------------------------------------------------------------------------
__global__ __launch_bounds__(256)
void capsule_em_routing_kernel(const float* __restrict__ v,
                               const float* __restrict__ a_in_g,
                               const float* __restrict__ beta_u,
                               const float* __restrict__ beta_a,
                               float* __restrict__ out) {
  __shared__ float s_r[NIN * CDIM];   // r / rr / coeff / ln_ap (36 KB)
  __shared__ float s_ain[NIN];
  __shared__ float s_mu[VCOLS];
  __shared__ float s_sig[VCOLS];
  __shared__ float s_rsum[CDIM];
  __shared__ float s_aout[CDIM];

  int tid = threadIdx.x;
  int n   = blockIdx.x;
  const float* vn = v + (size_t)n * NIN * VCOLS;

  for (int i = tid; i < NIN; i += 256) s_ain[i] = a_in_g[n * NIN + i];
  for (int idx = tid; idx < NIN * CDIM; idx += 256) s_r[idx] = 1.0f / CDIM;
  __syncthreads();

  for (int t = 0; t < EMITERS; ++t) {
    // rr = r * a_in, normalized over c (in place)
    for (int i = tid; i < NIN; i += 256) {
      float ai = s_ain[i];
      float sum = 0.f;
      for (int c = 0; c < CDIM; ++c) sum += s_r[i * CDIM + c] * ai;
      float inv = 1.f / (sum + EPS);
      for (int c = 0; c < CDIM; ++c) s_r[i * CDIM + c] *= ai * inv;
    }
    __syncthreads();
    // r_sum over input capsules
    if (tid < CDIM) {
      float s = 0.f;
      for (int i = 0; i < NIN; ++i) s += s_r[i * CDIM + tid];
      s_rsum[tid] = s;
    }
    __syncthreads();
    // coeff = rr / (r_sum + eps)  (in place)
    for (int idx = tid; idx < NIN * CDIM; idx += 256)
      s_r[idx] = s_r[idx] / (s_rsum[idx & 31] + EPS);
    __syncthreads();
    // mu
    for (int cq = tid; cq < VCOLS; cq += 256) {
      int c = cq >> 4;
      float acc = 0.f;
      for (int i = 0; i < NIN; ++i)
        acc += s_r[i * CDIM + c] * vn[i * VCOLS + cq];
      s_mu[cq] = acc;
    }
    __syncthreads();
    // sigma^2
    for (int cq = tid; cq < VCOLS; cq += 256) {
      int c = cq >> 4;
      float m = s_mu[cq];
      float acc = 0.f;
      for (int i = 0; i < NIN; ++i) {
        float d = vn[i * VCOLS + cq] - m;
        acc += s_r[i * CDIM + c] * d * d;
      }
      s_sig[cq] = fmaxf(acc + EPS, 1e-4f);
    }
    __syncthreads();
    // cost -> a_out
    if (tid < CDIM) {
      float bu = beta_u[tid];
      float cost = 0.f;
      for (int q = 0; q < PSIZE; ++q) {
        float sg = fmaxf(s_sig[tid * PSIZE + q], 1e-4f);
        cost += bu + 0.5f * logf(sg);
      }
      cost *= s_rsum[tid];
      float z = LAMBDA * (beta_a[tid] - cost);
      float ao = 1.f / (1.f + expf(-z));
      ao = fminf(fmaxf(ao, 1e-4f), 1.f - 1e-4f);
      if (isnan(ao)) ao = 0.5f;
      s_aout[tid] = ao;
    }
    __syncthreads();
    // nan_to_num on mu / sigma
    for (int cq = tid; cq < VCOLS; cq += 256) {
      float m = s_mu[cq];
      if (isnan(m)) m = 0.f;
      s_mu[cq] = fminf(fmaxf(m, -1e4f), 1e4f);
      float sg = s_sig[cq];
      if (isnan(sg)) sg = 1e-4f;
      s_sig[cq] = fminf(fmaxf(sg, 1e-4f), 1e4f);
    }
    __syncthreads();

    if (t < EMITERS - 1) {
      // ln_ap[i][c] into s_r
      for (int idx = tid; idx < NIN * CDIM; idx += 256) {
        int i = idx >> 5;
        int c = idx & 31;
        const float* vp = vn + i * VCOLS + c * PSIZE;
        float acc = 0.f;
#pragma unroll
        for (int q = 0; q < PSIZE; ++q) {
          float sg = s_sig[c * PSIZE + q];
          float d  = vp[q] - s_mu[c * PSIZE + q];
          acc += -d * d / (2.f * sg) - 0.5f * logf(sg) - 0.5f * LN_2PI;
        }
        acc += logf(s_aout[c] + EPS);
        if (isnan(acc)) acc = 0.f;
        s_r[idx] = fminf(fmaxf(acc, -50.f), 50.f);
      }
      __syncthreads();
      // softmax over c per input capsule
      for (int i = tid; i < NIN; i += 256) {
        float mx = -1e30f;
        for (int c = 0; c < CDIM; ++c) mx = fmaxf(mx, s_r[i * CDIM + c]);
        float ssum = 0.f;
        for (int c = 0; c < CDIM; ++c) ssum += expf(s_r[i * CDIM + c] - mx);
        float inv = 1.f / ssum;
        for (int c = 0; c < CDIM; ++c) {
          float rv = expf(s_r[i * CDIM + c] - mx) * inv;
          if (isnan(rv)) rv = 1.f / CDIM;
          s_r[i * CDIM + c] = rv;
        }
      }
      __syncthreads();
    }
  }

  // outputs: [p_out (128*512)] [a_out (128*32)] [concat (128*544)]
  const int OFF_A   = NPOS * VCOLS;            // 65536
  const int OFF_CAT = OFF_A + NPOS * CDIM;     // 69632
  for (int cq = tid; cq < VCOLS; cq += 256) {
    float m = s_mu[cq];                        // already nan_to_num'd
    out[n * VCOLS + cq] = m;
    out[OFF_CAT + n * 544 + cq] = m;
  }
  if (tid < CDIM) {
    float a = s_aout[tid];
    out[OFF_A + n * CDIM + tid] = a;
    out[OFF_CAT + n * 544 + VCOLS + tid] = a;
  }
}

// -------------------------------------------------------------------------
extern "C" void kernel_launch(void* const* d_in, const int* in_sizes, int n_in,
                              void* d_out, int out_size, void* d_ws, size_t ws_size,
                              hipStream_t stream) {
  (void)in_sizes; (void)n_in; (void)out_size; (void)ws_size;
  const float* x  = (const float*)d_in[0];
  const float* w  = (const float*)d_in[1];
  const float* bu = (const float*)d_in[2];
  const float* ba = (const float*)d_in[3];
  float* out = (float*)d_out;

  float* ws   = (float*)d_ws;
  float* p_in = ws;                                        // 589824 floats
  float* a_in = p_in + (size_t)NPOS * NIN * PSIZE;         // 36864 floats
  float* v    = a_in + (size_t)NPOS * NIN;                 // 18874368 floats (~75.5 MB)

  int total = NPOS * NIN * PSIZE + NPOS * NIN;
  capsule_extract_kernel<<<(total + 255) / 256, 256, 0, stream>>>(x, p_in, a_in);

  // 288 capsules * 8 M-tiles * 32 N-tiles = 73728 waves, 8 waves/block
  capsule_vote_wmma_kernel<<<(NIN * 8 * 32) / 8, 256, 0, stream>>>(w, p_in, v);

  capsule_em_routing_kernel<<<NPOS, 256, 0, stream>>>(v, a_in, bu, ba, out);
}